// TightBoxPrior_16784732193074
// MI455X (gfx1250) — compile-verified
//
#include <hip/hip_runtime.h>
#include <hip/hip_bf16.h>
#include <math.h>

typedef __attribute__((ext_vector_type(2))) float v2f;
typedef __attribute__((ext_vector_type(8))) float v8f;

#define HW    65536   // 256*256
#define NROWS 512     // N
#define NB    8       // B
#define TCONST 5.0f

// One workgroup per (batch, 16-row tile of N). 32 waves/block (wave32),
// each wave covers HW/32 = 2048 pixels with chained V_WMMA_F32_16X16X4_F32.
// A = masks rows (16 x 4 fp32), B = fg chunk broadcast across all 16 columns,
// so each column of the 16x16 accumulator holds the same partial dot product:
// lane 0 carries rows 0..7, lane 16 carries rows 8..15.
__global__ __launch_bounds__(1024) void tbp_sizes_kernel(
    const float* __restrict__ probs,   // (B, 2, H, W)
    const float* __restrict__ masks,   // (B, N, H, W)
    float* __restrict__ sizes)         // (B, N) in d_ws
{
  const int b    = blockIdx.x >> 5;          // 32 tiles per batch
  const int n0   = (blockIdx.x & 31) << 4;   // tile base row
  const int tid  = threadIdx.x;
  const int wave = tid >> 5;                 // 0..31
  const int lane = tid & 31;
  const int half = lane >> 4;                // 0: K=0,1   1: K=2,3
  const int row  = lane & 15;                // A-matrix M

  const float* fg   = probs + (size_t)b * 2 * HW + HW;              // probs[:,1]
  const float* mrow = masks + ((size_t)b * NROWS + (n0 + row)) * HW;

  const int hw0 = wave * (HW / 32);
  const int hw1 = hw0 + (HW / 32);

  v8f acc = {0.f, 0.f, 0.f, 0.f, 0.f, 0.f, 0.f, 0.f};

  for (int k = hw0; k < hw1; k += 32) {
    // Unguarded speculative prefetch ~2KB ahead of the masks stream
    // (global_prefetch_b8; OOB translations are silently dropped, no counters).
    __builtin_prefetch(mrow + k + 512, 0, 1);
#pragma unroll
    for (int u = 0; u < 8; ++u) {
      const int kk = k + u * 4 + half * 2;
      v2f a  = *(const v2f*)(mrow + kk);   // A 16x4 f32: VGPR0=K(0|2), VGPR1=K(1|3)
      v2f bb = *(const v2f*)(fg   + kk);   // B 4x16 f32 broadcast across columns
      acc = __builtin_amdgcn_wmma_f32_16x16x4_f32(
          /*neg_a=*/false, a, /*neg_b=*/false, bb,
          /*c_mod=*/(short)0, acc, /*reuse_a=*/false, /*reuse_b=*/false);
    }
  }

  // Cross-wave reduction: lane 0 has rows n0+0..7, lane 16 has rows n0+8..15.
  __shared__ float part[32][16];
  if (lane == 0) {
#pragma unroll
    for (int r = 0; r < 8; ++r) part[wave][r] = acc[r];
  } else if (lane == 16) {
#pragma unroll
    for (int r = 0; r < 8; ++r) part[wave][8 + r] = acc[r];
  }
  __syncthreads();

  if (tid < 16) {
    float s = 0.f;
#pragma unroll 8
    for (int w = 0; w < 32; ++w) s += part[w][tid];
    sizes[b * NROWS + n0 + tid] = s;
  }
}

// Elementwise error + cumprod(valid) + global sum on the tiny (8, 512) result.
// One lane per batch keeps the sequential cumprod semantics exactly.
__global__ void tbp_final_kernel(const float* __restrict__ bounds, // (B, N)
                                 const float* __restrict__ sizes,  // (B, N)
                                 float* __restrict__ out)
{
  __shared__ float acc[NB];
  const int b = threadIdx.x;
  const float ceil_v  = -1.0f / (TCONST * TCONST);
  const float b_const = logf(TCONST * TCONST) / TCONST + 1.0f / TCONST;

  if (b < NB) {
    float s = 0.f;
    bool valid = true;
    for (int n = 0; n < NROWS; ++n) {
      const float bd = bounds[b * NROWS + n];
      valid = valid && (bd != 0.f);           // cumprod of (bounds != 0)
      if (valid) {
        const float shifted = bd - sizes[b * NROWS + n];
        const float err = (shifted <= ceil_v)
                              ? (-logf(-shifted) / TCONST)
                              : (TCONST * shifted + b_const);
        s += err;
      }
    }
    acc[b] = s;
  }
  __syncthreads();

  if (threadIdx.x == 0) {
    float t = 0.f;
#pragma unroll
    for (int i = 0; i < NB; ++i) t += acc[i];
    out[0] = t / (float)NB;
  }
}

extern "C" void kernel_launch(void* const* d_in, const int* in_sizes, int n_in,
                              void* d_out, int out_size, void* d_ws, size_t ws_size,
                              hipStream_t stream) {
  const float* probs  = (const float*)d_in[0]; // (8, 2, 256, 256)
  const float* masks  = (const float*)d_in[1]; // (8, 512, 256, 256)
  const float* bounds = (const float*)d_in[2]; // (8, 512)
  float* sizes = (float*)d_ws;                 // 8*512 floats = 16 KB scratch
  float* out   = (float*)d_out;

  tbp_sizes_kernel<<<dim3(NB * (NROWS / 16)), dim3(1024), 0, stream>>>(probs, masks, sizes);
  tbp_final_kernel<<<dim3(1), dim3(32), 0, stream>>>(bounds, sizes, out);
}